// mamba2DuelingModel_78391743086956
// MI455X (gfx1250) — compile-verified
//
#include <hip/hip_runtime.h>
#include <math.h>

// ---------------- model dims ----------------
#define BATCH   64
#define SEQ     1024
#define DIN     32
#define HID     64
#define DINNER  128
#define DSTATE  128
#define NH      2
#define PDIM    64          // HEADDIM
#define CONVDIM 384
#define DPROJ   514         // 2*DINNER + 2*DSTATE + NH
#define TOK     (BATCH*SEQ) // 65536

typedef __attribute__((ext_vector_type(2))) float v2f;
typedef __attribute__((ext_vector_type(8))) float v8f;

static __device__ __forceinline__ float siluf(float v) { return v / (1.f + __expf(-v)); }
static __device__ __forceinline__ float sigm(float v)  { return 1.f / (1.f + __expf(-v)); }
static __device__ __forceinline__ float wave_sum(float v) {
  v += __shfl_xor(v, 1, 32);
  v += __shfl_xor(v, 2, 32);
  v += __shfl_xor(v, 4, 32);
  v += __shfl_xor(v, 8, 32);
  v += __shfl_xor(v, 16, 32);
  return v;
}

// ================= generic fp32 WMMA GEMM =================
// C[M,N] = act(A[M,K] @ B[K,N] + bias)   (mode 0)
// C[M,N] += A@B                          (mode 1: acc init from C -> residual add)
// atomicAdd(C, partial A@B)              (mode 2: split-K, C must be zeroed)
// One wave computes one 16x16 tile via V_WMMA_F32_16X16X4_F32.
// Requires K % 4 == 0 and kChunk % 4 == 0 (true for every launch below).
// Out-of-range lanes are CLAMPED to valid rows/cols (not predicated): an A-row
// only feeds its own D-row and a B-col its own D-col, and those D elements are
// masked at the store, so the K-loop carries no EXEC manipulation at all.
__global__ void gemm_wmma_f32(const float* __restrict__ A, const float* __restrict__ Bw,
                              const float* __restrict__ bias, float* __restrict__ C,
                              int M, int N, int K, int lda, int ldb, int ldc,
                              int kChunk, int mode, int act)
{
  const int wave = threadIdx.x >> 5;
  const int lane = threadIdx.x & 31;
  const int half = lane >> 4;   // 0: lanes 0-15, 1: lanes 16-31
  const int lm   = lane & 15;
  const int tilesN = (N + 15) >> 4;
  const int tilesM = (M + 15) >> 4;
  const int tileId = blockIdx.x * 8 + wave;
  if (tileId >= tilesM * tilesN) return;           // wave-uniform: EXEC stays full
  const int m0 = (tileId / tilesN) << 4;
  const int n0 = (tileId % tilesN) << 4;
  const int k0 = blockIdx.y * kChunk;
  int kEnd = k0 + kChunk; if (kEnd > K) kEnd = K;

  const int  mA  = m0 + lm;
  const int  nB  = n0 + lm;
  const bool mok = (mA < M);
  const bool nok = (nB < N);
  const int  mAc = mok ? mA : m0;   // clamped (m0 always < M, n0 always < N)
  const int  nBc = nok ? nB : n0;

  const size_t ldbz = (size_t)ldb;
  const float* Ap = A + (size_t)mAc * lda + k0 + half * 2;      // 8B aligned (lda, k0 even)
  const float* Bp = Bw + ((size_t)k0 + half * 2) * ldbz + nBc;

  v8f acc, acc2;
  if (mode == 1) {
#pragma unroll
    for (int r = 0; r < 8; ++r) {
      const int m = m0 + r + half * 8;             // C/D layout: vgpr r -> M=r (+8 hi half)
      acc[r] = (m < M && nok) ? C[(size_t)m * ldc + nB] : 0.f;
    }
  } else {
#pragma unroll
    for (int r = 0; r < 8; ++r) acc[r] = 0.f;
  }
#pragma unroll
  for (int r = 0; r < 8; ++r) acc2[r] = 0.f;

  int nsteps = (kEnd - k0) >> 2;                   // number of K=4 WMMA steps
  // two independent accumulator chains: 2 WMMAs per iteration overlap with loads
  while (nsteps >= 2) {
    const v2f a0 = *(const v2f*)Ap;                // global_load_b64
    const v2f a1 = *(const v2f*)(Ap + 4);
    v2f b0, b1;
    b0.x = Bp[0];
    b0.y = Bp[ldbz];
    b1.x = Bp[4 * ldbz];
    b1.y = Bp[5 * ldbz];
    acc  = __builtin_amdgcn_wmma_f32_16x16x4_f32(false, a0, false, b0, (short)0, acc,  false, false);
    acc2 = __builtin_amdgcn_wmma_f32_16x16x4_f32(false, a1, false, b1, (short)0, acc2, false, false);
    Ap += 8;
    Bp += 8 * ldbz;
    nsteps -= 2;
  }
  if (nsteps) {                                    // tail (K % 8 == 4)
    const v2f a0 = *(const v2f*)Ap;
    v2f b0;
    b0.x = Bp[0];
    b0.y = Bp[ldbz];
    acc = __builtin_amdgcn_wmma_f32_16x16x4_f32(false, a0, false, b0, (short)0, acc, false, false);
  }
#pragma unroll
  for (int r = 0; r < 8; ++r) acc[r] += acc2[r];

#pragma unroll
  for (int r = 0; r < 8; ++r) {
    const int m = m0 + r + half * 8;
    if (m < M && nok) {
      float v = acc[r];
      if (mode == 2) {
        atomicAdd(&C[(size_t)m * ldc + nB], v);
      } else {
        if (bias) v += bias[nB];
        if (act == 1)      v = fmaxf(v, 0.f);
        else if (act == 2) v = tanhf(v);
        C[(size_t)m * ldc + nB] = v;
      }
    }
  }
}

// ================= elementwise / norm kernels =================

__global__ void zero_kernel(float* p, int n) {
  int i = blockIdx.x * 256 + threadIdx.x;
  if (i < n) p[i] = 0.f;
}

// rms over rows of width W (32 lanes per row, W/32 elems each)
__global__ void rmsnorm_kernel(const float* __restrict__ in, const float* __restrict__ w,
                               float* __restrict__ out, int rows, int W)
{
  const int wave = threadIdx.x >> 5, lane = threadIdx.x & 31;
  const int row = blockIdx.x * 8 + wave;
  if (row >= rows) return;
  const int per = W >> 5;
  const float* src = in + (size_t)row * W;
  float v[4]; float ss = 0.f;
  for (int i = 0; i < per; ++i) { v[i] = src[lane + i * 32]; ss += v[i] * v[i]; }
  ss = wave_sum(ss);
  const float sc = rsqrtf(ss / (float)W + 1e-5f);
  float* dst = out + (size_t)row * W;
  for (int i = 0; i < per; ++i) dst[lane + i * 32] = v[i] * sc * w[lane + i * 32];
}

// dt = softplus(zx[...,512+h] + dt_bias) ; dA = exp(dt * -exp(A_log))
__global__ void dt_da_kernel(const float* __restrict__ zx, const float* __restrict__ dt_bias,
                             const float* __restrict__ A_log, float* __restrict__ dtO,
                             float* __restrict__ dAO)
{
  int i = blockIdx.x * 256 + threadIdx.x;   // over TOK*NH
  if (i >= TOK * NH) return;
  int row = i >> 1, hh = i & 1;
  float v = zx[(size_t)row * DPROJ + 512 + hh] + dt_bias[hh];
  float sp = (v > 20.f) ? v : log1pf(__expf(v));
  float An = -__expf(A_log[hh]);
  dtO[i] = sp;
  dAO[i] = __expf(sp * An);
}

// SSD scan: one block per (batch, head). Depthwise causal conv1d + SiLU computed
// on the fly from a 4-deep LDS ring of raw zx columns; state (64x128) in VGPRs.
__global__ void __launch_bounds__(256) scan_kernel(
    const float* __restrict__ zx, const float* __restrict__ conv_w,
    const float* __restrict__ conv_b, const float* __restrict__ dt,
    const float* __restrict__ dA, const float* __restrict__ Dskip,
    float* __restrict__ y)
{
  const int b    = blockIdx.x >> 1;
  const int head = blockIdx.x & 1;
  const int tid  = threadIdx.x;

  __shared__ float ring[4][320];   // raw xBC cols, times t-3..t
  __shared__ float cur[320];       // conv+silu output at t: [0,64)=xs(head) [64,192)=B [192,320)=C

  // channel map: slot j -> conv channel c ; zx column = 128 + c
  const int j1 = tid;                                 // 0..255
  const int c1 = (j1 < 64) ? head * 64 + j1 : j1 + 64;
  const int j2 = tid + 256;                           // 256..319 when tid < 64
  const int c2 = j2 + 64;

  float w1r[4], w2r[4] = {0.f, 0.f, 0.f, 0.f}, b1r, b2r = 0.f;
#pragma unroll
  for (int k = 0; k < 4; ++k) w1r[k] = conv_w[c1 * 4 + k];
  b1r = conv_b[c1];
  if (tid < 64) {
#pragma unroll
    for (int k = 0; k < 4; ++k) w2r[k] = conv_w[c2 * 4 + k];
    b2r = conv_b[c2];
  }
#pragma unroll
  for (int s = 0; s < 4; ++s) { ring[s][j1] = 0.f; if (tid < 64) ring[s][j2] = 0.f; }

  float h[32];
#pragma unroll
  for (int j = 0; j < 32; ++j) h[j] = 0.f;

  const int p  = tid >> 2;          // headdim element 0..63
  const int nb = (tid & 3) << 5;    // state slice base 0/32/64/96
  const float dsk = Dskip[head];

  for (int t = 0; t < SEQ; ++t) {
    const size_t row = (size_t)b * SEQ + t;
    __syncthreads();                               // prior conv reads done before overwrite
    ring[t & 3][j1] = zx[row * DPROJ + 128 + c1];
    if (tid < 64) ring[t & 3][j2] = zx[row * DPROJ + 128 + c2];
    if (t + 1 < SEQ && (tid & 15) == 0)            // hint next column (global_prefetch_b8)
      __builtin_prefetch(&zx[(row + 1) * DPROJ + 128 + c1], 0, 1);
    __syncthreads();
    {                                              // causal conv (k taps at t-3..t) + SiLU
      float s1 = b1r;
#pragma unroll
      for (int k = 0; k < 4; ++k) s1 += w1r[k] * ring[(t + 1 + k) & 3][j1];
      cur[j1] = siluf(s1);
      if (tid < 64) {
        float s2 = b2r;
#pragma unroll
        for (int k = 0; k < 4; ++k) s2 += w2r[k] * ring[(t + 1 + k) & 3][j2];
        cur[j2] = siluf(s2);
      }
    }
    __syncthreads();
    const float dtv = dt[row * NH + head];
    const float dAv = dA[row * NH + head];
    const float xv  = cur[p];
    const float dtx = dtv * xv;
    float acc = 0.f;
#pragma unroll
    for (int j = 0; j < 32; ++j) {
      const float bj = cur[64 + nb + j];
      const float cj = cur[192 + nb + j];
      h[j] = h[j] * dAv + dtx * bj;
      acc += h[j] * cj;
    }
    acc += __shfl_xor(acc, 1, 32);                 // reduce the 4 n-slices of this p
    acc += __shfl_xor(acc, 2, 32);
    if ((tid & 3) == 0)
      y[row * DINNER + head * 64 + p] = acc + dsk * xv;
  }
}

// y <- rms(y * silu(z), norm_g), in place; z = zx cols [0,128)
__global__ void gated_rmsnorm_kernel(float* __restrict__ y, const float* __restrict__ zx,
                                     const float* __restrict__ g, int rows)
{
  const int wave = threadIdx.x >> 5, lane = threadIdx.x & 31;
  const int row = blockIdx.x * 8 + wave;
  if (row >= rows) return;
  float v[4]; float ss = 0.f;
  for (int i = 0; i < 4; ++i) {
    const int c = lane + i * 32;
    const float zv = zx[(size_t)row * DPROJ + c];
    const float yv = y[(size_t)row * DINNER + c];
    v[i] = yv * siluf(zv);
    ss += v[i] * v[i];
  }
  ss = wave_sum(ss);
  const float sc = rsqrtf(ss / (float)DINNER + 1e-5f);
  for (int i = 0; i < 4; ++i) {
    const int c = lane + i * 32;
    y[(size_t)row * DINNER + c] = v[i] * sc * g[c];
  }
}

// fused gated MLP (d_inter=1): x += (rms(x,n2)@w1 gate) @ w2   (wave per token)
__global__ void mlp_kernel(float* __restrict__ x, const float* __restrict__ n2,
                           const float* __restrict__ w1, const float* __restrict__ w2, int rows)
{
  const int wave = threadIdx.x >> 5, lane = threadIdx.x & 31;
  const int row = blockIdx.x * 8 + wave;
  if (row >= rows) return;
  float* xr = x + (size_t)row * HID;
  const float x0 = xr[lane], x1 = xr[lane + 32];
  float ss = wave_sum(x0 * x0 + x1 * x1);
  const float sc = rsqrtf(ss / (float)HID + 1e-5f);
  const float m0 = x0 * sc * n2[lane], m1 = x1 * sc * n2[lane + 32];
  float yv = wave_sum(m0 * w1[lane * 2 + 0] + m1 * w1[(lane + 32) * 2 + 0]);
  float gv = wave_sum(m0 * w1[lane * 2 + 1] + m1 * w1[(lane + 32) * 2 + 1]);
  const float s = yv * siluf(gv);
  xr[lane]      = x0 + s * w2[lane];
  xr[lane + 32] = x1 + s * w2[lane + 32];
}

// LayerNorm(pre + bias)*g + bb, ReLU, in place; one block per row of width W (<=512)
__global__ void ln_relu_kernel(float* __restrict__ h, const float* __restrict__ bias,
                               const float* __restrict__ g, const float* __restrict__ bb, int W)
{
  const int row = blockIdx.x, tid = threadIdx.x;
  __shared__ float red[256];
  __shared__ float stat[2];
  const int per = W >> 8;   // 2 for 512, 1 for 256
  float v[2] = {0.f, 0.f};
  float s = 0.f;
  for (int i = 0; i < per; ++i) {
    const int c = tid + i * 256;
    v[i] = h[(size_t)row * W + c] + bias[c];
    s += v[i];
  }
  red[tid] = s; __syncthreads();
  for (int o = 128; o > 0; o >>= 1) { if (tid < o) red[tid] += red[tid + o]; __syncthreads(); }
  if (tid == 0) stat[0] = red[0] / (float)W;
  __syncthreads();
  const float mean = stat[0];
  float s2 = 0.f;
  for (int i = 0; i < per; ++i) { const float d = v[i] - mean; s2 += d * d; }
  red[tid] = s2; __syncthreads();
  for (int o = 128; o > 0; o >>= 1) { if (tid < o) red[tid] += red[tid + o]; __syncthreads(); }
  if (tid == 0) stat[1] = red[0] / (float)W;
  __syncthreads();
  const float rstd = rsqrtf(stat[1] + 1e-5f);
  for (int i = 0; i < per; ++i) {
    const int c = tid + i * 256;
    h[(size_t)row * W + c] = fmaxf(0.f, (v[i] - mean) * rstd * g[c] + bb[c]);
  }
}

// ================= DAIN =================
// block per (b,d); mode 0: mean over L ; mode 1: mean of (src-adj)^2
__global__ void dain_reduce(const float* __restrict__ src, const float* __restrict__ adj,
                            float* __restrict__ out, int mode)
{
  const int bd = blockIdx.x, b = bd >> 5, d = bd & 31, tid = threadIdx.x;
  __shared__ float red[256];
  const float a = (mode == 1) ? adj[bd] : 0.f;
  float s = 0.f;
  for (int l = tid; l < SEQ; l += 256) {
    float v = src[((size_t)b * SEQ + l) * DIN + d] - a;
    s += (mode == 1) ? v * v : v;
  }
  red[tid] = s; __syncthreads();
  for (int o = 128; o > 0; o >>= 1) { if (tid < o) red[tid] += red[tid + o]; __syncthreads(); }
  if (tid == 0) out[bd] = red[0] * (1.f / (float)SEQ);
}

__global__ void dain_small1(const float* __restrict__ avg, const float* __restrict__ mw,
                            float* __restrict__ adj)
{
  const int bd = blockIdx.x * 256 + threadIdx.x;
  if (bd >= BATCH * DIN) return;
  const int b = bd >> 5, d = bd & 31;
  float s = 0.f;
  for (int k = 0; k < 32; ++k) s += avg[b * 32 + k] * mw[k * 32 + d];
  adj[bd] = s;
}

__global__ void dain_small2(const float* __restrict__ ss, const float* __restrict__ avg,
                            const float* __restrict__ adj, const float* __restrict__ sw,
                            float* __restrict__ m2, float* __restrict__ inva)
{
  const int bd = blockIdx.x * 256 + threadIdx.x;
  if (bd >= BATCH * DIN) return;
  const int b = bd >> 5, d = bd & 31;
  float s = 0.f;
  for (int k = 0; k < 32; ++k) s += sqrtf(ss[b * 32 + k] + 1e-8f) * sw[k * 32 + d];
  if (s <= 1e-8f) s = 1.f;
  const float ia = 1.f / s;
  inva[bd] = ia;
  m2[bd] = (avg[bd] - adj[bd]) * ia;
}

__global__ void dain_small3(const float* __restrict__ m2, const float* __restrict__ gw,
                            const float* __restrict__ gb, const float* __restrict__ inva,
                            const float* __restrict__ adj, float* __restrict__ sc,
                            float* __restrict__ sh)
{
  const int bd = blockIdx.x * 256 + threadIdx.x;
  if (bd >= BATCH * DIN) return;
  const int b = bd >> 5, d = bd & 31;
  float s = 0.f;
  for (int k = 0; k < 32; ++k) s += m2[b * 32 + k] * gw[k * 32 + d];
  const float gate = sigm(s + gb[d]);
  const float scv = gate * inva[bd];
  sc[bd] = scv;
  sh[bd] = -adj[bd] * scv;
}

__global__ void dain_apply(const float* __restrict__ src, const float* __restrict__ sc,
                           const float* __restrict__ sh, float* __restrict__ xd)
{
  const int i = blockIdx.x * 256 + threadIdx.x;
  if (i >= TOK * DIN) return;
  const int b = i >> 15;           // / (SEQ*DIN)
  const int d = i & 31;
  xd[i] = src[i] * sc[b * 32 + d] + sh[b * 32 + d];
}

// ================= dueling combine =================
__global__ void head_out_kernel(const float* __restrict__ h2v, const float* __restrict__ h2a,
                                const float* __restrict__ w3v, const float* __restrict__ b3v,
                                const float* __restrict__ w3a, const float* __restrict__ b3a,
                                float* __restrict__ out)
{
  __shared__ float vs[64];
  __shared__ float as[64][3];
  const int tid = threadIdx.x, b = tid >> 2, r = tid & 3;
  float s = 0.f;
  if (r == 0) {
    for (int k = 0; k < 256; ++k) s += h2v[b * 256 + k] * w3v[k];
    vs[b] = s + b3v[0];
  } else {
    const int j = r - 1;
    for (int k = 0; k < 256; ++k) s += h2a[b * 256 + k] * w3a[k * 3 + j];
    as[b][j] = s + b3a[j];
  }
  __syncthreads();
  if (r) {
    const float m = (as[b][0] + as[b][1] + as[b][2]) * (1.f / 3.f);
    out[b * 3 + (r - 1)] = vs[b] + as[b][r - 1] - m;
  }
}

// ================= host orchestration =================
static void launch_gemm(const float* A, const float* Bw, const float* bias, float* C,
                        int M, int N, int K, int lda, int ldb, int ldc,
                        int splitK, int mode, int act, hipStream_t s)
{
  const int tiles = ((M + 15) / 16) * ((N + 15) / 16);
  int kChunk = (K + splitK - 1) / splitK;
  kChunk = (kChunk + 3) & ~3;
  dim3 grid((tiles + 7) / 8, splitK);
  gemm_wmma_f32<<<grid, 256, 0, s>>>(A, Bw, bias, C, M, N, K, lda, ldb, ldc, kChunk, mode, act);
}

// workspace layout (floats) -- total ~212 MB
static const size_t OFF_XD  = 0;                         // TOK*32
static const size_t OFF_X   = OFF_XD  + (size_t)TOK * DIN;      // TOK*64
static const size_t OFF_XN  = OFF_X   + (size_t)TOK * HID;      // TOK*64
static const size_t OFF_ZX  = OFF_XN  + (size_t)TOK * HID;      // TOK*514
static const size_t OFF_DT  = OFF_ZX  + (size_t)TOK * DPROJ;    // TOK*2
static const size_t OFF_DA  = OFF_DT  + (size_t)TOK * NH;       // TOK*2
static const size_t OFF_Y   = OFF_DA  + (size_t)TOK * NH;       // TOK*128
static const size_t OFF_AVG = OFF_Y   + (size_t)TOK * DINNER;
static const size_t OFF_ADJ = OFF_AVG + 2048;
static const size_t OFF_SS  = OFF_ADJ + 2048;
static const size_t OFF_M2  = OFF_SS  + 2048;
static const size_t OFF_IA  = OFF_M2  + 2048;
static const size_t OFF_SC  = OFF_IA  + 2048;
static const size_t OFF_SH  = OFF_SC  + 2048;
static const size_t OFF_H1V = OFF_SH  + 2048;            // 64*512
static const size_t OFF_H1A = OFF_H1V + 64 * 512;
static const size_t OFF_H2V = OFF_H1A + 64 * 512;        // 64*256
static const size_t OFF_H2A = OFF_H2V + 64 * 256;

extern "C" void kernel_launch(void* const* d_in, const int* in_sizes, int n_in,
                              void* d_out, int out_size, void* d_ws, size_t ws_size,
                              hipStream_t stream)
{
  (void)in_sizes; (void)n_in; (void)out_size; (void)ws_size;
#define IN(i) ((const float*)d_in[(i)])
  const float* src = IN(0);
  // dain: 1..4   embed: 5..8   layers: 9..104 (12 each)   norm_f: 105   heads: 106..125
  float* ws   = (float*)d_ws;
  float* xd   = ws + OFF_XD;
  float* x    = ws + OFF_X;
  float* xn   = ws + OFF_XN;
  float* zx   = ws + OFF_ZX;
  float* dtb  = ws + OFF_DT;
  float* dab  = ws + OFF_DA;
  float* yb   = ws + OFF_Y;
  float* avg  = ws + OFF_AVG;
  float* adj  = ws + OFF_ADJ;
  float* ssb  = ws + OFF_SS;
  float* m2   = ws + OFF_M2;
  float* inva = ws + OFF_IA;
  float* scb  = ws + OFF_SC;
  float* shb  = ws + OFF_SH;

  // ---- DAIN (affine per (b,d): out = src*sc + sh) ----
  dain_reduce<<<BATCH * DIN, 256, 0, stream>>>(src, nullptr, avg, 0);
  dain_small1<<<8, 256, 0, stream>>>(avg, IN(1), adj);
  dain_reduce<<<BATCH * DIN, 256, 0, stream>>>(src, adj, ssb, 1);
  dain_small2<<<8, 256, 0, stream>>>(ssb, avg, adj, IN(2), m2, inva);
  dain_small3<<<8, 256, 0, stream>>>(m2, IN(3), IN(4), inva, adj, scb, shb);
  dain_apply<<<(TOK * DIN + 255) / 256, 256, 0, stream>>>(src, scb, shb, xd);

  // ---- embed: tanh(relu(xd@w1+b1)@w2+b2) ----
  launch_gemm(xd, IN(5), IN(6), xn, TOK, HID, DIN, DIN, HID, HID, 1, 0, 1, stream); // relu
  launch_gemm(xn, IN(7), IN(8), x,  TOK, HID, HID, HID, HID, HID, 1, 0, 2, stream); // tanh

  // ---- 8 Mamba2 layers ----
  for (int l = 0; l < 8; ++l) {
    const int L0 = 9 + 12 * l;
    const float* p_norm1 = IN(L0 + 0);
    const float* p_inprj = IN(L0 + 1);
    const float* p_convw = IN(L0 + 2);
    const float* p_convb = IN(L0 + 3);
    const float* p_dtb   = IN(L0 + 4);
    const float* p_Alog  = IN(L0 + 5);
    const float* p_Dsk   = IN(L0 + 6);
    const float* p_normg = IN(L0 + 7);
    const float* p_outp  = IN(L0 + 8);
    const float* p_norm2 = IN(L0 + 9);
    const float* p_mw1   = IN(L0 + 10);
    const float* p_mw2   = IN(L0 + 11);

    rmsnorm_kernel<<<TOK / 8, 256, 0, stream>>>(x, p_norm1, xn, TOK, HID);
    launch_gemm(xn, p_inprj, nullptr, zx, TOK, DPROJ, HID, HID, DPROJ, DPROJ, 1, 0, 0, stream);
    dt_da_kernel<<<(TOK * NH + 255) / 256, 256, 0, stream>>>(zx, p_dtb, p_Alog, dtb, dab);
    scan_kernel<<<BATCH * NH, 256, 0, stream>>>(zx, p_convw, p_convb, dtb, dab, p_Dsk, yb);
    gated_rmsnorm_kernel<<<TOK / 8, 256, 0, stream>>>(yb, zx, p_normg, TOK);
    // residual: x += yn @ out_proj  (mode 1 initializes accumulator from x)
    launch_gemm(yb, p_outp, nullptr, x, TOK, HID, DINNER, DINNER, HID, HID, 1, 1, 0, stream);
    mlp_kernel<<<TOK / 8, 256, 0, stream>>>(x, p_norm2, p_mw1, p_mw2, TOK);
  }

  // ---- final norm; view as (B, L*H) ----
  rmsnorm_kernel<<<TOK / 8, 256, 0, stream>>>(x, IN(105), xn, TOK, HID);

  // ---- dueling heads (split-K WMMA for the 64x65536x512 GEMM) ----
  float* h1[2] = { ws + OFF_H1V, ws + OFF_H1A };
  float* h2[2] = { ws + OFF_H2V, ws + OFF_H2A };
  for (int hd = 0; hd < 2; ++hd) {
    const int HB = 106 + 10 * hd;   // w1,b1,g1,bb1,w2,b2,g2,bb2,w3,b3
    zero_kernel<<<(64 * 512 + 255) / 256, 256, 0, stream>>>(h1[hd], 64 * 512);
    launch_gemm(xn, IN(HB + 0), nullptr, h1[hd], BATCH, 512, SEQ * HID,
                SEQ * HID, 512, 512, 64, 2, 0, stream);          // split-K=64, atomic
    ln_relu_kernel<<<BATCH, 256, 0, stream>>>(h1[hd], IN(HB + 1), IN(HB + 2), IN(HB + 3), 512);
    launch_gemm(h1[hd], IN(HB + 4), nullptr, h2[hd], BATCH, 256, 512, 512, 256, 256,
                1, 0, 0, stream);
    ln_relu_kernel<<<BATCH, 256, 0, stream>>>(h2[hd], IN(HB + 5), IN(HB + 6), IN(HB + 7), 256);
  }
  head_out_kernel<<<1, 256, 0, stream>>>(h2[0], h2[1], IN(106 + 8), IN(106 + 9),
                                         IN(116 + 8), IN(116 + 9), (float*)d_out);
#undef IN
}